// PointPillarScatter_77713138254101
// MI455X (gfx1250) — compile-verified
//
#include <hip/hip_runtime.h>

// PointPillars KITTI scatter: [B*P, C] features -> dense [B, C, NY, NX] canvas.
// Pure data movement (0 FLOPs). Optimized as an inverted gather with
// non-temporal b128 store streaming (gfx1250 TH=NT cache hint).

typedef float v4f __attribute__((ext_vector_type(4)));
typedef int   v4i __attribute__((ext_vector_type(4)));

#define PP_NX    432
#define PP_NY    496
#define PP_C     64
#define PP_B     8
#define PP_GRID  (PP_NX * PP_NY)         // 214272 (divisible by 4)
#define PP_BGRID (PP_B * PP_GRID)        // 1714176
#define MAP_G4   (PP_BGRID / 4)          // 428544 int4 groups (= 1674 * 256)
#define PLANE_G4 (PP_GRID / 4)           // 53568 groups per (b) plane

// ---------------------------------------------------------------------------
// Kernel 1: fill pillar-index map with -1 (vectorized 16B stores).
// ---------------------------------------------------------------------------
__global__ void pp_fill_map(v4i* __restrict__ map4) {
    int i = blockIdx.x * blockDim.x + threadIdx.x;   // exact grid, no guard
    v4i m = {-1, -1, -1, -1};
    map4[i] = m;
}

// ---------------------------------------------------------------------------
// Kernel 2: scatter pillar ids into the map. coords row = [b, z, y, x].
// gidx = b*GRID + z + y*NX + x   (matches reference exactly).
// ---------------------------------------------------------------------------
__global__ void pp_scatter_idx(const v4i* __restrict__ coords4,
                               int* __restrict__ map, int P) {
    int p = blockIdx.x * blockDim.x + threadIdx.x;
    if (p >= P) return;
    v4i c = coords4[p];
    int g = c.x * PP_GRID + c.y + c.z * PP_NX + c.w;
    map[g] = p;
}

// ---------------------------------------------------------------------------
// Kernel 3: output-coalesced gather. Each thread owns 4 consecutive grid
// cells of one sample; loops over 64 channels emitting one NT float4 store
// per channel plane (global_store_b128 th:NT -> bypass/stream past L2 so the
// 8.4MB feature table + 6.9MB map stay L2-resident).
// ---------------------------------------------------------------------------
__global__ void pp_gather(const float* __restrict__ feat,
                          const v4i* __restrict__ map4,
                          float* __restrict__ out) {
    int gg = blockIdx.x * blockDim.x + threadIdx.x;  // group-of-4 index
    v4i pv = map4[gg];                               // coalesced 16B map read
    int b  = gg / PLANE_G4;                          // groups never straddle b
    int g0 = (gg - b * PLANE_G4) << 2;
    float* outb = out + (size_t)b * (PP_C * PP_GRID) + g0;

    // All four cells empty <=> AND of sign bits set. ~98% of threads take
    // this path: pure zero streaming, no feature reads at all.
    if ((pv.x & pv.y & pv.z & pv.w) < 0) {
        v4f z = {0.f, 0.f, 0.f, 0.f};
#pragma unroll 8
        for (int c = 0; c < PP_C; ++c)
            __builtin_nontemporal_store(z, (v4f*)(outb + (size_t)c * PP_GRID));
    } else {
#pragma unroll 4
        for (int c = 0; c < PP_C; ++c) {
            v4f v;
            v.x = (pv.x >= 0) ? feat[(size_t)pv.x * PP_C + c] : 0.f;
            v.y = (pv.y >= 0) ? feat[(size_t)pv.y * PP_C + c] : 0.f;
            v.z = (pv.z >= 0) ? feat[(size_t)pv.z * PP_C + c] : 0.f;
            v.w = (pv.w >= 0) ? feat[(size_t)pv.w * PP_C + c] : 0.f;
            __builtin_nontemporal_store(v, (v4f*)(outb + (size_t)c * PP_GRID));
        }
    }
}

// ---------------------------------------------------------------------------
// Fallback path (only if ws_size < 6.9MB): zero-fill + direct scatter.
// ---------------------------------------------------------------------------
__global__ void pp_zero(v4f* __restrict__ out4, int n4) {
    int i = blockIdx.x * blockDim.x + threadIdx.x;
    if (i >= n4) return;
    v4f z = {0.f, 0.f, 0.f, 0.f};
    __builtin_nontemporal_store(z, out4 + i);
}

__global__ void pp_scatter_direct(const float* __restrict__ feat,
                                  const int* __restrict__ coords,
                                  float* __restrict__ out, int P) {
    int t = blockIdx.x * blockDim.x + threadIdx.x;   // t = p*C + c
    if (t >= P * PP_C) return;
    int p = t >> 6;
    int c = t & (PP_C - 1);
    int b = coords[p * 4 + 0];
    int z = coords[p * 4 + 1];
    int y = coords[p * 4 + 2];
    int x = coords[p * 4 + 3];
    size_t o = (size_t)b * (PP_C * PP_GRID) + (size_t)c * PP_GRID
             + (size_t)(z + y * PP_NX + x);
    out[o] = feat[t];                                // coalesced read, scattered write
}

// ---------------------------------------------------------------------------
extern "C" void kernel_launch(void* const* d_in, const int* in_sizes, int n_in,
                              void* d_out, int out_size, void* d_ws, size_t ws_size,
                              hipStream_t stream) {
    const float* feat   = (const float*)d_in[0];   // [P, 64] fp32
    const int*   coords = (const int*)d_in[1];     // [P, 4]  int32
    float*       out    = (float*)d_out;           // [8, 64, 496, 432] fp32

    const int P = in_sizes[0] / PP_C;              // 32768
    const size_t map_bytes = (size_t)PP_BGRID * sizeof(int);   // ~6.9 MB

    if (ws_size >= map_bytes) {
        int* map = (int*)d_ws;
        pp_fill_map   <<<MAP_G4 / 256, 256, 0, stream>>>((v4i*)map);
        pp_scatter_idx<<<(P + 255) / 256, 256, 0, stream>>>((const v4i*)coords, map, P);
        pp_gather     <<<MAP_G4 / 256, 256, 0, stream>>>(feat, (const v4i*)map, out);
    } else {
        const int n4 = (int)((size_t)PP_B * PP_C * PP_GRID / 4);  // 27,426,816
        pp_zero          <<<(n4 + 255) / 256, 256, 0, stream>>>((v4f*)out, n4);
        pp_scatter_direct<<<(P * PP_C + 255) / 256, 256, 0, stream>>>(feat, coords, out, P);
    }
}